// CenterLoss_80015240724894
// MI455X (gfx1250) — compile-verified
//
#include <hip/hip_runtime.h>

#define NUM_CLASSES 100000
#define FEAT_DIM    256
#define BATCH       16384
#define REG_ALPHA   0.5f
#define REG_LAMBDA  1.0f
#define ROWS_PER_BLOCK 8   // 8 rows * 256 f32 = 8 KB LDS tile

typedef __attribute__((ext_vector_type(2))) float v2f;
typedef __attribute__((ext_vector_type(8))) float v8f;

// ---- CDNA5 async global->LDS copy (CDNA5 ISA) ----
__device__ __forceinline__ void async_load_b128(unsigned lds_byte_addr, const void* gptr) {
  // VDST = LDS byte address (VGPR), VADDR = 64-bit global address (VGPR pair), GV mode
  asm volatile("global_load_async_to_lds_b128 %0, %1, off"
               :: "v"(lds_byte_addr), "v"(gptr)
               : "memory");
}
__device__ __forceinline__ void wait_asynccnt0() {
  asm volatile("s_wait_asynccnt 0" ::: "memory");
}

// ---------------------------------------------------------------------------
// Kernel 0: zero only the accumulator rows that this batch will touch
// (~16 MB of coalesced b32 stores instead of a 102 MB memset; duplicate
// classes just re-write zero, which is idempotent).
// One thread per (batch row, feature) element: 16384*256 threads.
// ---------------------------------------------------------------------------
__global__ void __launch_bounds__(256)
centerloss_zero_touched(const int* __restrict__ targets,
                        float*     __restrict__ acc)   // d_out+1
{
  const int gid = blockIdx.x * 256 + threadIdx.x;
  const int i = gid >> 8;    // batch row (all 256 threads of a block share it)
  const int d = gid & 255;   // feature index
  const int t = targets[i];
  acc[(size_t)t * FEAT_DIM + d] = 0.0f;
}

// ---------------------------------------------------------------------------
// Kernel 1: gather + squared-error partials + scatter-add of (c - f) and counts.
// One thread per float4 of the (B, D) matrix: 16384*64 = 1,048,576 threads.
// Loss wave-reduction done with V_WMMA_F32_16X16X4_F32 (B = ones => row sums).
// ---------------------------------------------------------------------------
__global__ void __launch_bounds__(256)
centerloss_scatter(const float* __restrict__ features,
                   const int*   __restrict__ targets,
                   const float* __restrict__ centers,
                   float*       __restrict__ acc,        // d_out+1, touched rows zeroed
                   float*       __restrict__ counts,     // ws, zeroed
                   float*       __restrict__ loss_accum) // ws, zeroed
{
  const int gid = blockIdx.x * 256 + threadIdx.x;
  const int i = gid >> 6;   // batch row
  const int v = gid & 63;   // float4 column within row

  const int t = targets[i];
  const float4 f = ((const float4*)features)[(size_t)i * 64 + v];
  const float4 c = ((const float4*)centers )[(size_t)t * 64 + v];

  float4 d;
  d.x = c.x - f.x;
  d.y = c.y - f.y;
  d.z = c.z - f.z;
  d.w = c.w - f.w;

  float* ap = acc + (size_t)t * FEAT_DIM + (size_t)v * 4;
  atomicAdd(ap + 0, d.x);
  atomicAdd(ap + 1, d.y);
  atomicAdd(ap + 2, d.z);
  atomicAdd(ap + 3, d.w);
  if (v == 0) atomicAdd(&counts[t], 1.0f);

  // per-thread squared-error partial; (f-c)^2 == d^2
  float s = d.x * d.x + d.y * d.y + d.z * d.z + d.w * d.w;

  // Wave32 reduction via WMMA: A(16x4) holds each lane's partial at
  // (M = lane%16, K = lane<16 ? 0 : 2); B = ones(4x16) => D[m][n] = rowsum(m).
  // Summing a lane's 8 D-VGPRs gives sum over M=0..7 (lanes 0-15) or
  // M=8..15 (lanes 16-31); lane0 + lane16 = full 32-lane total.
  v2f a; a[0] = s;    a[1] = 0.0f;
  v2f b; b[0] = 1.0f; b[1] = 1.0f;
  v8f cm = {};
  cm = __builtin_amdgcn_wmma_f32_16x16x4_f32(
      /*neg_a=*/false, a, /*neg_b=*/false, b,
      /*c_mod=*/(short)0, cm, /*reuse_a=*/false, /*reuse_b=*/false);
  float lanesum = cm[0] + cm[1] + cm[2] + cm[3] + cm[4] + cm[5] + cm[6] + cm[7];

  const unsigned lane = threadIdx.x & 31u;
  if (lane == 0u || lane == 16u) atomicAdd(loss_accum, lanesum);
}

// ---------------------------------------------------------------------------
// Kernel 2: new_centers = centers - alpha * acc / (counts + 1) for touched
// rows, new_centers = centers for untouched rows (never reads their stale
// accumulator), in place over d_out[1:]. Streams `centers` tiles through LDS
// with CDNA5 async b128 loads. Also finalizes the scalar loss into d_out[0].
// ---------------------------------------------------------------------------
__global__ void __launch_bounds__(256)
centerloss_finalize(const float* __restrict__ centers,
                    const float* __restrict__ counts,
                    const float* __restrict__ loss_accum,
                    float*       __restrict__ out)  // d_out
{
  __shared__ alignas(16) float tile[ROWS_PER_BLOCK * FEAT_DIM]; // 8 KB

  const int tid = threadIdx.x;
  const size_t row0 = (size_t)blockIdx.x * ROWS_PER_BLOCK;
  const size_t base = row0 * FEAT_DIM;                          // element base

  // Stage 8 KB of centers into LDS: 512 x b128, 2 per thread.
  const unsigned lds_base = (unsigned)(uintptr_t)(&tile[0]);
#pragma unroll
  for (int u = 0; u < 2; ++u) {
    const unsigned byte_off = (unsigned)(tid + u * 256) * 16u;
    async_load_b128(lds_base + byte_off,
                    (const void*)((const char*)(centers + base) + byte_off));
  }
  wait_asynccnt0();
  __syncthreads();

  float* __restrict__ acc = out + 1; // 4B-aligned only -> scalar b32 traffic
#pragma unroll
  for (int e = 0; e < ROWS_PER_BLOCK; ++e) {
    const int ti = tid + e * FEAT_DIM;     // row e, column tid of the tile
    const size_t gi = base + (size_t)ti;
    const float cnt = counts[row0 + e];    // uniform across the row
    float val = tile[ti];
    if (cnt > 0.0f) {                      // wave-uniform branch
      val -= REG_ALPHA * acc[gi] / (cnt + 1.0f);
    }
    acc[gi] = val;                         // write new_centers in place
  }

  if (blockIdx.x == 0 && tid == 0) {
    out[0] = loss_accum[0] * (REG_LAMBDA / (float)((size_t)BATCH * FEAT_DIM));
  }
}

extern "C" void kernel_launch(void* const* d_in, const int* in_sizes, int n_in,
                              void* d_out, int out_size, void* d_ws, size_t ws_size,
                              hipStream_t stream) {
  (void)in_sizes; (void)n_in; (void)out_size; (void)ws_size;

  const float* features = (const float*)d_in[0];
  const int*   targets  = (const int*)  d_in[1];
  const float* centers  = (const float*)d_in[2];

  float* out        = (float*)d_out;           // [0] loss, [1:] new_centers
  float* counts     = (float*)d_ws;            // NUM_CLASSES floats
  float* loss_accum = counts + NUM_CLASSES;    // 1 float

  // Zero the small accumulators every call (graph-capture-safe, deterministic).
  hipMemsetAsync(d_ws, 0, (size_t)(NUM_CLASSES + 1) * sizeof(float), stream);

  // Zero only the delta-accumulator rows this batch touches (16384 blocks).
  centerloss_zero_touched<<<BATCH, 256, 0, stream>>>(targets, out + 1);

  // (BATCH * FEAT_DIM / 4) threads = 1,048,576 -> 4096 blocks of 256 (no tail).
  centerloss_scatter<<<(BATCH * (FEAT_DIM / 4)) / 256, 256, 0, stream>>>(
      features, targets, centers, out + 1, counts, loss_accum);

  // 100000 / 8 = 12500 blocks (exact).
  centerloss_finalize<<<NUM_CLASSES / ROWS_PER_BLOCK, 256, 0, stream>>>(
      centers, counts, loss_accum, out);
}